// NodeEmbeddingNetwork2_42863773614330
// MI455X (gfx1250) — compile-verified
//
#include <hip/hip_runtime.h>
#include <stdint.h>

// Embedding gather:  out[b,c,n,f] = W[f, idx[b,c,n]] + bias[f]
//   B=32, C=16, N=2048, F=64, VOCAB=4096  -> out = 256 MB f32 (bandwidth bound)
//
// Roofline: 256 MB output + 4 MB idx + 1 MB table ≈ 11 us at 23.3 TB/s HBM.
// WMMA is deliberately NOT used: one-hot @ W as dense matmul = ~550 GFLOP of
// wasted work vs. an HBM-limited copy.
//
// CDNA5-native structure:
//   1) prep_table: Wt[v][f] = W[f][v] + bias[f]  (1 MB, bias fused, rows = 256 B
//      contiguous; stays resident in the 192 MB L2 across 256x reuse)
//   2) gather_async: per-wave double-buffered DMA pipeline using the gfx1250
//      async path — GLOBAL_LOAD_ASYNC_TO_LDS_B128 gathers table rows into LDS,
//      GLOBAL_STORE_ASYNC_FROM_LDS_B128 (th:TH_STORE_NT) streams them out.
//      Data never touches the VGPR file; ASYNCcnt tracks both directions.
//      In flight per wave between waits: 4 KB stores + 4 KB loads; x8192 waves
//      = 64 MB outstanding >> 16 MB BW*latency product -> HBM saturated.

typedef float v4f __attribute__((ext_vector_type(4)));

#define EMB_VOCAB 4096
#define EMB_F     64
#define EMB_TOKENS (32 * 16 * 2048)      // 1,048,576 tokens, 256 B each

#define TOK_PER_BATCH   16               // 16 tokens * 256 B = 4 KB per batch
#define OPS_PER_BATCH   8                // 8 async ops * (32 lanes * 16 B) = 4 KB
#define BUF_BYTES       4096             // one batch buffer
#define WAVES_PER_BLOCK 8                // 256 threads
#define GATHER_BLOCKS   1024
#define TOTAL_WAVES     (GATHER_BLOCKS * WAVES_PER_BLOCK)          // 8192
#define TOTAL_BATCHES   (EMB_TOKENS / TOK_PER_BATCH)               // 65536

// ---------------------------------------------------------------------------
// Pass 1: bias-fused transpose  Wt[v][f] = W[f][v] + bias[f]
// ---------------------------------------------------------------------------
__global__ void NodeEmb_prep_table(const float* __restrict__ W,
                                   const float* __restrict__ bias,
                                   float* __restrict__ Wt) {
    int t = blockIdx.x * blockDim.x + threadIdx.x;   // 0 .. VOCAB*F-1
    int v = t >> 6;
    int f = t & (EMB_F - 1);
    Wt[t] = W[f * EMB_VOCAB + v] + bias[f];
}

// ---------------------------------------------------------------------------
// Async DMA helpers (gfx1250): LDS byte offsets are relative to the block's
// LDS allocation base (hardware adds LDS_BASE on both load-to-LDS and
// store-from-LDS, so raw offsets into the sole __shared__ array are correct).
// ---------------------------------------------------------------------------
__device__ __forceinline__ void async_gather_batch(const int* __restrict__ idx,
                                                   const float* __restrict__ Wt,
                                                   int tokBase, uint32_t ldsBase,
                                                   int lane) {
    // Lanes 0..15 fetch the batch's 16 indices coalesced (lanes 16..31 merge
    // into the same cacheline request); broadcast per-op via wave32 shuffle.
    int myIdx = __builtin_nontemporal_load(&idx[tokBase + (lane & 15)]);
    const int comp = lane & 15;                    // which float4 of the row
    const uint64_t wtB = (uint64_t)(uintptr_t)Wt + (uint32_t)(comp << 4);
#pragma unroll
    for (int j = 0; j < OPS_PER_BATCH; ++j) {
        int tok = 2 * j + (lane >> 4);             // token within batch, 0..15
        int v   = __shfl(myIdx, tok, 32);          // ds_bpermute broadcast
        uint64_t ga = wtB + ((uint64_t)(uint32_t)v << 8);     // row = 256 B
        uint32_t lo = ldsBase + (uint32_t)(j * 512 + lane * 16);
        asm volatile("global_load_async_to_lds_b128 %0, %1, off"
                     :: "v"(lo), "v"(ga) : "memory");
    }
}

__device__ __forceinline__ void async_store_batch(float* __restrict__ out,
                                                  int tokBase, uint32_t ldsBase,
                                                  int lane) {
    const uint64_t outB = (uint64_t)(uintptr_t)out + (uint64_t)tokBase * 256u;
#pragma unroll
    for (int j = 0; j < OPS_PER_BATCH; ++j) {
        uint64_t ga = outB + (uint32_t)((j * 32 + lane) * 16); // contiguous 4 KB
        uint32_t lo = ldsBase + (uint32_t)(j * 512 + lane * 16);
        // NT: 256 MB touch-once stream must not evict the L2-resident table.
        asm volatile("global_store_async_from_lds_b128 %0, %1, off th:TH_STORE_NT"
                     :: "v"(ga), "v"(lo) : "memory");
    }
}

__device__ __forceinline__ void wait_async0() {
    asm volatile("s_wait_asynccnt 0" ::: "memory");
}

// ---------------------------------------------------------------------------
// Pass 2: double-buffered async gather pipeline.
// ---------------------------------------------------------------------------
__global__ void NodeEmb_gather_async(const int* __restrict__ idx,
                                     const float* __restrict__ Wt,
                                     float* __restrict__ out) {
    __shared__ uint8_t smem[WAVES_PER_BLOCK * 2 * BUF_BYTES];   // 64 KB / block

    const int lane = threadIdx.x & 31;
    const int wave = threadIdx.x >> 5;

    // Touch LDS with a plain (ds) store so the allocation is concrete, then
    // fence it against the async engine before DMA writes land in it.
    ((volatile uint32_t*)smem)[threadIdx.x] = 0u;
    __syncthreads();

    const uint32_t ldsWave = (uint32_t)(wave * 2 * BUF_BYTES);
    const int gwave = blockIdx.x * WAVES_PER_BLOCK + wave;

    uint32_t buf = 0;
    int b = gwave;
    if (b < TOTAL_BATCHES) {
        // Prologue: start loads for this wave's first batch.
        async_gather_batch(idx, Wt, b * TOK_PER_BATCH, ldsWave, lane);
    }
    for (; b < TOTAL_BATCHES; b += TOTAL_WAVES) {
        // All outstanding async ops done here: loads of batch b (buf) and
        // stores of the previous batch (buf^1). So buf's data is ready to
        // stream out, and buf^1 is free for the next batch's loads.
        wait_async0();
        async_store_batch(out, b * TOK_PER_BATCH, ldsWave + buf * BUF_BYTES, lane);
        int nb = b + TOTAL_WAVES;
        if (nb < TOTAL_BATCHES) {
            async_gather_batch(idx, Wt, nb * TOK_PER_BATCH,
                               ldsWave + (buf ^ 1u) * BUF_BYTES, lane);
        }
        buf ^= 1u;
    }
    wait_async0();   // drain final stores (S_ENDPGM also implies wait-idle)
}

// ---------------------------------------------------------------------------
extern "C" void kernel_launch(void* const* d_in, const int* in_sizes, int n_in,
                              void* d_out, int out_size, void* d_ws, size_t ws_size,
                              hipStream_t stream) {
    const int*   x    = (const int*)d_in[0];    // [B, C, 1, N] int32 ids
    const float* W    = (const float*)d_in[1];  // [F, VOCAB] f32
    const float* bias = (const float*)d_in[2];  // [F] f32

    float* Wt = (float*)d_ws;                   // [VOCAB, F] f32 = 1 MB scratch

    // Pass 1: bias-fused transposed table (1 MB; negligible cost).
    NodeEmb_prep_table<<<(EMB_VOCAB * EMB_F) / 256, 256, 0, stream>>>(W, bias, Wt);

    // Pass 2: async-DMA gather, 1024 blocks x 8 waves, 8 batches per wave.
    NodeEmb_gather_async<<<GATHER_BLOCKS, WAVES_PER_BLOCK * 32, 0, stream>>>(
        x, Wt, (float*)d_out);
}